// GraphSageNNV4_88132728914382
// MI455X (gfx1250) — compile-verified
//
#include <hip/hip_runtime.h>
#include <hip/hip_bf16.h>

typedef __attribute__((ext_vector_type(2))) float v2f;
typedef __attribute__((ext_vector_type(8))) float v8f;

constexpr int D_IN = 128, D_HID = 256, D_OUT = 64;

// ---------------------------------------------------------------- zero fill
__global__ void zero4_kernel(float4* __restrict__ p, long long n4) {
    long long i = (long long)blockIdx.x * blockDim.x + threadIdx.x;
    long long stride = (long long)gridDim.x * blockDim.x;
    float4 z = make_float4(0.f, 0.f, 0.f, 0.f);
    for (; i < n4; i += stride) p[i] = z;
}

// ---------------------------------------------------------------- in-degree
__global__ void count_deg_kernel(const int* __restrict__ dst, float* __restrict__ deg, int E) {
    int e = blockIdx.x * blockDim.x + threadIdx.x;
    if (e < E) atomicAdd(&deg[dst[e]], 1.0f);
}

// ------------------------------------------------- gather + scatter-add (sum)
// One wave32 per edge; lane moves float4 chunks. agg buffer is L2-resident
// (102 MB < 192 MB L2) so the f32 atomics resolve on-chip.
template <int D>
__global__ void scatter_add_kernel(const float* __restrict__ feat,
                                   const int* __restrict__ src,
                                   const int* __restrict__ dst,
                                   float* __restrict__ agg, int E) {
    int wave = threadIdx.x >> 5;
    int lane = threadIdx.x & 31;
    int e = blockIdx.x * (blockDim.x >> 5) + wave;
    if (e >= E) return;
    int s = src[e], d = dst[e];
    const float* fp = feat + (size_t)s * D;
    float* ap = agg + (size_t)d * D;
#pragma unroll
    for (int j = 0; j < D / 128; ++j) {
        int c = 4 * lane + 128 * j;
        float4 v = *(const float4*)(fp + c);
        atomicAdd(ap + c + 0, v.x);
        atomicAdd(ap + c + 1, v.y);
        atomicAdd(ap + c + 2, v.z);
        atomicAdd(ap + c + 3, v.w);
    }
}

// ------------------------------------------------------------- fused SAGE GEMM
// out[m][n] = act( (A[m][:] * invdeg[m]) . W[n][:] + bias[n] )
// One wave per 16x16 tile, fp32 WMMA 16x16x4, K-loop of K/4 steps.
// Fragment layouts per CDNA5 ISA 7.12.2:
//   A f32 16x4 : v_j, lane-half hf -> (M = lane&15, K = j + 2*hf)
//   B f32 4x16 : v_j, lane-half hf -> (K = j + 2*hf, N = lane&15)
//   C/D f32    : v_r, lane-half hf -> (M = r + 8*hf, N = lane&15)
template <int K, int NCOLS, bool RELU>
__global__ void sage_gemm_kernel(const float* __restrict__ A,
                                 const float* __restrict__ deg,
                                 const float* __restrict__ W,   // [NCOLS, K] row-major (PyTorch W)
                                 const float* __restrict__ bias,
                                 float* __restrict__ out) {
    constexpr int NTN = NCOLS / 16;
    int wave = threadIdx.x >> 5;
    int lane = threadIdx.x & 31;
    int tile = blockIdx.x * (blockDim.x >> 5) + wave;
    int tm = tile / NTN;
    int tn = tile % NTN;
    int hf = lane >> 4;
    int l  = lane & 15;

    int row = tm * 16 + l;   // A row held by this lane
    int col = tn * 16 + l;   // W row (= output column) held by this lane
    float invd = 1.0f / fmaxf(deg[row], 1.0f);   // scatter-mean normalization, fused

    const float* ap = A + (size_t)row * K + 2 * hf;
    const float* bp = W + (size_t)col * K + 2 * hf;

    v8f acc = {};
#pragma unroll 4
    for (int kk = 0; kk < K / 4; ++kk) {
        float2 av = *(const float2*)(ap + 4 * kk);
        float2 bv = *(const float2*)(bp + 4 * kk);
        v2f a; a[0] = av.x * invd; a[1] = av.y * invd;
        v2f b; b[0] = bv.x;        b[1] = bv.y;
        acc = __builtin_amdgcn_wmma_f32_16x16x4_f32(
            /*neg_a=*/false, a, /*neg_b=*/false, b,
            /*c_mod=*/(short)0, acc, /*reuse_a=*/false, /*reuse_b=*/false);
    }

    float bb = bias[col];
#pragma unroll
    for (int r = 0; r < 8; ++r) {
        int m = tm * 16 + r + 8 * hf;
        float v = acc[r] + bb;
        if (RELU) v = fmaxf(v, 0.0f);
        out[(size_t)m * NCOLS + col] = v;
    }
}

// ---------------------------------------------------------- log_softmax (64)
// One wave32 per row; float2 per lane; wave32 shuffle reductions.
__global__ void log_softmax64_kernel(float* __restrict__ y, int N) {
    int wave = threadIdx.x >> 5;
    int lane = threadIdx.x & 31;
    int rowid = blockIdx.x * (blockDim.x >> 5) + wave;
    if (rowid >= N) return;
    float* p = y + (size_t)rowid * 64;
    float2 v = *(const float2*)(p + 2 * lane);
    float m = fmaxf(v.x, v.y);
#pragma unroll
    for (int off = 16; off > 0; off >>= 1) m = fmaxf(m, __shfl_xor(m, off, 32));
    float s = expf(v.x - m) + expf(v.y - m);
#pragma unroll
    for (int off = 16; off > 0; off >>= 1) s += __shfl_xor(s, off, 32);
    float lg = m + logf(s);
    float2 o = make_float2(v.x - lg, v.y - lg);
    *(float2*)(p + 2 * lane) = o;
}

// ---------------------------------------------------------------- launcher
extern "C" void kernel_launch(void* const* d_in, const int* in_sizes, int n_in,
                              void* d_out, int out_size, void* d_ws, size_t ws_size,
                              hipStream_t stream) {
    const float* x  = (const float*)d_in[0];
    const int*   ei = (const int*)d_in[1];   // [2, E]: src row then dst row
    const float* W1 = (const float*)d_in[2]; // [256, 128]
    const float* b1 = (const float*)d_in[3];
    const float* W2 = (const float*)d_in[4]; // [64, 256]
    const float* b2 = (const float*)d_in[5];
    float* out = (float*)d_out;              // [N, 64] log-probs

    const int N = in_sizes[0] / D_IN;        // 100000 (divisible by 16)
    const int E = in_sizes[1] / 2;           // 1600000 (divisible by 8)
    const int* src = ei;
    const int* dst = ei + E;

    // workspace layout: deg | agg (N*256, shared by both layers) | h (N*256)
    const int Npad = (N + 63) & ~63;
    float* deg = (float*)d_ws;
    float* agg = deg + Npad;
    float* h   = agg + (size_t)N * D_HID;

    // --- layer 1 ---
    {   // zero deg + agg[0 : N*128)
        long long n4 = ((long long)Npad + (long long)N * D_IN) / 4;
        zero4_kernel<<<2048, 256, 0, stream>>>((float4*)deg, n4);
    }
    count_deg_kernel<<<(E + 255) / 256, 256, 0, stream>>>(dst, deg, E);
    scatter_add_kernel<D_IN><<<(E + 7) / 8, 256, 0, stream>>>(x, src, dst, agg, E);
    sage_gemm_kernel<D_IN, D_HID, true>
        <<<(N / 16) * (D_HID / 16) / 8, 256, 0, stream>>>(agg, deg, W1, b1, h);

    // --- layer 2 ---
    zero4_kernel<<<4096, 256, 0, stream>>>((float4*)agg, (long long)N * D_HID / 4);
    scatter_add_kernel<D_HID><<<(E + 7) / 8, 256, 0, stream>>>(h, src, dst, agg, E);
    sage_gemm_kernel<D_HID, D_OUT, false>
        <<<(N / 16) * (D_OUT / 16) / 8, 256, 0, stream>>>(agg, deg, W2, b2, out);

    log_softmax64_kernel<<<(N + 7) / 8, 256, 0, stream>>>(out, N);
}